// OriginCEALNetwork_70695161692649
// MI455X (gfx1250) — compile-verified
//
#include <hip/hip_runtime.h>
#include <math.h>

typedef float v2f __attribute__((ext_vector_type(2)));
typedef float v8f __attribute__((ext_vector_type(8)));

#define AVG_LOG 1.9595436f

// ---------------- WMMA f32 16x16x4 helpers ----------------
__device__ __forceinline__ v8f wmma4(v2f a, v2f b, v8f c) {
  // D = A(16x4) * B(4x16) + C, full f32
  return __builtin_amdgcn_wmma_f32_16x16x4_f32(false, a, false, b, (short)0, c, false, false);
}

// A tile is row-major [16 x lda], fragment for K-step k.
// ISA layout: VGPR0 = K (lanes0-15) / K+2 (lanes16-31); VGPR1 = K+1 / K+3.
__device__ __forceinline__ v2f afrag(const float* A, int lda, int k, int lane) {
  int r  = lane & 15;
  int kk = k + ((lane >> 4) << 1);
  v2f a; a.x = A[r * lda + kk]; a.y = A[r * lda + kk + 1]; return a;
}
// B is row-major [K x ldb]; fragment rows k..k+3, cols col0..col0+15.
__device__ __forceinline__ v2f bfrag(const float* B, int ldb, int k, int col0, int lane) {
  int c  = col0 + (lane & 15);
  int kk = k + ((lane >> 4) << 1);
  v2f b; b.x = B[kk * ldb + c]; b.y = B[(kk + 1) * ldb + c]; return b;
}
// D/C layout: lane<16 -> col, rows 0..7 ; lane>=16 -> col, rows 8..15
__device__ __forceinline__ void store_d(float* D, int ldd, int col0, v8f d, int lane) {
  int c  = col0 + (lane & 15);
  int r0 = (lane >> 4) << 3;
#pragma unroll
  for (int i = 0; i < 8; ++i) D[(size_t)(r0 + i) * ldd + c] = d[i];
}

// ---------------- weight prep: Wec = We @ Wp2, cvec = be @ Wp2 ----------------
__global__ void k_prep(const float* __restrict__ We, const float* __restrict__ be,
                       const float* __restrict__ Wp, float* __restrict__ Wec,
                       float* __restrict__ cvec) {
  int j = threadIdx.x;  // 64 threads
  float cv = 0.f;
  for (int i = 0; i < 64; ++i) cv += be[i] * Wp[(128 + i) * 64 + j];
  cvec[j] = cv;
  for (int k = 0; k < 16; ++k) {
    float acc = 0.f;
    for (int i = 0; i < 64; ++i) acc += We[k * 64 + i] * Wp[(128 + i) * 64 + j];
    Wec[k * 64 + j] = acc;
  }
}

// ---------------- init accumulators ----------------
__global__ void k_init(float* sum, float* sumsq, unsigned* mne, unsigned* mxe,
                       float* cnt, float* colsum, float* colsumsq, float* pooled,
                       int n64, int n) {
  int i = blockIdx.x * blockDim.x + threadIdx.x;
  if (i < n64) { sum[i] = 0.f; sumsq[i] = 0.f; mne[i] = 0xFFFFFFFFu; mxe[i] = 0u; }
  if (i < n) cnt[i] = 0.f;
  if (i < 64) { colsum[i] = 0.f; colsumsq[i] = 0.f; }
  if (i < 64 * 64) pooled[i] = 0.f;
}

// ---------------- pre-MLP: xnew = relu(x@W1+b1)@W2+b2 ----------------
__global__ __launch_bounds__(128) void k_premlp(
    const float* __restrict__ x, const float* __restrict__ W1, const float* __restrict__ b1,
    const float* __restrict__ W2, const float* __restrict__ b2,
    float* __restrict__ xnew, int ntiles) {
  __shared__ float sW1[64 * 32], sW2[32 * 64], sb1[32], sb2[64];
  __shared__ float sX[4][16 * 64];
  __shared__ float sH[4][16 * 32];
  int tid = threadIdx.x, lane = tid & 31, w = tid >> 5;
  for (int i = tid; i < 64 * 32; i += 128) sW1[i] = W1[i];
  for (int i = tid; i < 32 * 64; i += 128) sW2[i] = W2[i];
  if (tid < 32) sb1[tid] = b1[tid];
  if (tid < 64) sb2[tid] = b2[tid];
  int t = blockIdx.x * 4 + w;
  bool active = t < ntiles;
  if (active) {
    const float* xt = x + (size_t)t * 16 * 64;
    for (int i = lane; i < 16 * 64; i += 32) sX[w][i] = xt[i];
  }
  __syncthreads();
  if (active) {
    for (int nt = 0; nt < 2; ++nt) {
      v8f acc; float bv = sb1[nt * 16 + (lane & 15)];
#pragma unroll
      for (int i = 0; i < 8; ++i) acc[i] = bv;
      for (int k = 0; k < 64; k += 4)
        acc = wmma4(afrag(sX[w], 64, k, lane), bfrag(sW1, 32, k, nt * 16, lane), acc);
#pragma unroll
      for (int i = 0; i < 8; ++i) acc[i] = fmaxf(acc[i], 0.f);
      store_d(sH[w], 32, nt * 16, acc, lane);
    }
  }
  __syncthreads();
  if (active) {
    float* ot = xnew + (size_t)t * 16 * 64;
    for (int nt = 0; nt < 4; ++nt) {
      v8f acc; float bv = sb2[nt * 16 + (lane & 15)];
#pragma unroll
      for (int i = 0; i < 8; ++i) acc[i] = bv;
      for (int k = 0; k < 32; k += 4)
        acc = wmma4(afrag(sH[w], 32, k, lane), bfrag(sW2, 64, k, nt * 16, lane), acc);
      store_d(ot, 64, nt * 16, acc, lane);
    }
  }
}

// ---------------- projections: xd = xnew@Wp[0:64], xs = xnew@Wp[64:128] ----------------
__global__ __launch_bounds__(128) void k_proj(
    const float* __restrict__ xnew, const float* __restrict__ Wp,
    float* __restrict__ xd, float* __restrict__ xs, int ntiles) {
  __shared__ float sWa[64 * 64], sWb[64 * 64];
  __shared__ float sX[4][16 * 64];
  int tid = threadIdx.x, lane = tid & 31, w = tid >> 5;
  for (int i = tid; i < 64 * 64; i += 128) { sWa[i] = Wp[i]; sWb[i] = Wp[4096 + i]; }
  int t = blockIdx.x * 4 + w;
  bool active = t < ntiles;
  if (active) {
    const float* xt = xnew + (size_t)t * 16 * 64;
    for (int i = lane; i < 16 * 64; i += 32) sX[w][i] = xt[i];
  }
  __syncthreads();
  if (active) {
    for (int which = 0; which < 2; ++which) {
      const float* Wm = which ? sWb : sWa;
      float* out = (which ? xs : xd) + (size_t)t * 16 * 64;
      for (int nt = 0; nt < 4; ++nt) {
        v8f acc;
#pragma unroll
        for (int i = 0; i < 8; ++i) acc[i] = 0.f;
        for (int k = 0; k < 64; k += 4)
          acc = wmma4(afrag(sX[w], 64, k, lane), bfrag(Wm, 64, k, nt * 16, lane), acc);
        store_d(out, 64, nt * 16, acc, lane);
      }
    }
  }
}

// ---------------- edge scatter: m = xs[src] + eattr@Wec + cvec ; reduce at dst ----------------
__global__ __launch_bounds__(256) void k_scatter(
    const float* __restrict__ edge_attr, const int* __restrict__ eidx,
    const float* __restrict__ xs, const float* __restrict__ Wec,
    const float* __restrict__ cvec, float* __restrict__ sum, float* __restrict__ sumsq,
    unsigned* __restrict__ mne, unsigned* __restrict__ mxe, float* __restrict__ cnt, int E) {
  __shared__ float sWec[16 * 64];
  __shared__ float sCv[64];
  int tid = threadIdx.x, lane = tid & 31, w = tid >> 5;
  for (int i = tid; i < 1024; i += 256) sWec[i] = Wec[i];
  if (tid < 64) sCv[tid] = cvec[tid];
  __syncthreads();
  int e = blockIdx.x * 8 + w;
  if (e >= E) return;
  int src = eidx[e];
  int dst = eidx[E + e];
  float eav = (lane < 16) ? edge_attr[(size_t)e * 16 + lane] : 0.f;
  int j0 = lane, j1 = lane + 32;
  float m0 = sCv[j0], m1 = sCv[j1];
#pragma unroll
  for (int k = 0; k < 16; ++k) {
    float ek = __shfl(eav, k, 32);
    m0 += ek * sWec[k * 64 + j0];
    m1 += ek * sWec[k * 64 + j1];
  }
  size_t sb = (size_t)src * 64;
  m0 += xs[sb + j0]; m1 += xs[sb + j1];
  size_t db = (size_t)dst * 64;
  atomicAdd(&sum[db + j0], m0);           atomicAdd(&sum[db + j1], m1);
  atomicAdd(&sumsq[db + j0], m0 * m0);    atomicAdd(&sumsq[db + j1], m1 * m1);
  unsigned u0 = __float_as_uint(m0); u0 = (u0 & 0x80000000u) ? ~u0 : (u0 | 0x80000000u);
  unsigned u1 = __float_as_uint(m1); u1 = (u1 & 0x80000000u) ? ~u1 : (u1 | 0x80000000u);
  atomicMin(&mne[db + j0], u0); atomicMin(&mne[db + j1], u1);
  atomicMax(&mxe[db + j0], u0); atomicMax(&mxe[db + j1], u1);
  if (lane == 0) atomicAdd(&cnt[dst], 1.0f);
}

__device__ __forceinline__ float dec_ord(unsigned u) {
  u = (u & 0x80000000u) ? (u & 0x7FFFFFFFu) : ~u;
  return __uint_as_float(u);
}

// ---------------- node post: out2 = (cat[x,scaled]@Wo+bo)@Wl+bl, plus BN col stats ----------------
__global__ __launch_bounds__(128) void k_node(
    const float* __restrict__ xnew, const float* __restrict__ xd,
    const float* __restrict__ sum, const float* __restrict__ sumsq,
    const unsigned* __restrict__ mne, const unsigned* __restrict__ mxe,
    const float* __restrict__ cnt, const float* __restrict__ bp,
    const float* __restrict__ Wo, const float* __restrict__ bo,
    const float* __restrict__ Wl, const float* __restrict__ bl,
    float* __restrict__ out2, float* __restrict__ colsum, float* __restrict__ colsumsq,
    int ntiles) {
  __shared__ float sAgg[4][16 * 256];   // per wave: [row][mean|mn|mx|std]
  __shared__ float sX[4][16 * 64];      // x tile; reused as out tile in phase 3
  __shared__ float sAmp[4][16], sInv[4][16];
  __shared__ float sCS[64], sCS2[64];
  int tid = threadIdx.x, lane = tid & 31, w = tid >> 5;
  if (tid < 64) { sCS[tid] = 0.f; sCS2[tid] = 0.f; }
  int t = blockIdx.x * 4 + w;
  bool active = t < ntiles;
  int r0 = t * 16;
  if (active) {
    for (int i = lane; i < 1024; i += 32) {
      int r = i >> 6, c = i & 63;
      size_t g = (size_t)(r0 + r) * 64 + c;
      float cv = cnt[r0 + r];
      float d = fmaxf(cv, 1.f);
      float s1 = sum[g], s2 = sumsq[g];
      float mm = s1 / d;
      float var = s2 / d - mm * mm;
      float sd = sqrtf(fmaxf(var, 0.f) + 1e-5f);      // std is shift-invariant
      float base = xd[g] + bp[c];                     // xd[dst]+bp shift term
      bool has = cv > 0.f;
      sAgg[w][r * 256 + c]        = has ? base + mm : 0.f;
      sAgg[w][r * 256 + 64 + c]   = has ? base + dec_ord(mne[g]) : 0.f;
      sAgg[w][r * 256 + 128 + c]  = has ? base + dec_ord(mxe[g]) : 0.f;
      sAgg[w][r * 256 + 192 + c]  = sd;
      sX[w][r * 64 + c] = xnew[g];
    }
    if (lane < 16) {
      float d = fmaxf(cnt[r0 + lane], 1.f);
      float amp = logf(d + 1.f) / AVG_LOG;
      sAmp[w][lane] = amp; sInv[w][lane] = 1.f / amp;
    }
  }
  __syncthreads();
  v8f acc[4];
  if (active) {
    for (int nt = 0; nt < 4; ++nt) {
      float bv = bo[nt * 16 + (lane & 15)];
#pragma unroll
      for (int i = 0; i < 8; ++i) acc[nt][i] = bv;
    }
    // segment 0: x @ Wo[0:64]
    for (int k = 0; k < 64; k += 4) {
      v2f a = afrag(sX[w], 64, k, lane);
      for (int nt = 0; nt < 4; ++nt)
        acc[nt] = wmma4(a, bfrag(Wo, 64, k, nt * 16, lane), acc[nt]);
    }
    // segments 1..12: {mean,mn,mx,std} x {1, amp, 1/amp}; scaler folded into A rows
    int r = lane & 15;
    for (int s = 1; s < 13; ++s) {
      int ai = (s - 1) & 3, sc = (s - 1) >> 2;
      float fr = (sc == 0) ? 1.f : ((sc == 1) ? sAmp[w][r] : sInv[w][r]);
      const float* arow = &sAgg[w][r * 256 + ai * 64];
      const float* Bo = Wo + (size_t)s * 64 * 64;
      for (int k = 0; k < 64; k += 4) {
        int kk = k + ((lane >> 4) << 1);
        v2f a; a.x = arow[kk] * fr; a.y = arow[kk + 1] * fr;
        for (int nt = 0; nt < 4; ++nt)
          acc[nt] = wmma4(a, bfrag(Bo, 64, k, nt * 16, lane), acc[nt]);
      }
    }
  }
  __syncthreads();
  if (active)
    for (int nt = 0; nt < 4; ++nt) store_d(sX[w], 64, nt * 16, acc[nt], lane);
  __syncthreads();
  if (active) {
    float* ot = out2 + (size_t)t * 16 * 64;
    for (int nt = 0; nt < 4; ++nt) {
      v8f a2; float bv = bl[nt * 16 + (lane & 15)];
#pragma unroll
      for (int i = 0; i < 8; ++i) a2[i] = bv;
      for (int k = 0; k < 64; k += 4)
        a2 = wmma4(afrag(sX[w], 64, k, lane), bfrag(Wl, 64, k, nt * 16, lane), a2);
      store_d(ot, 64, nt * 16, a2, lane);
      float p = 0.f, p2 = 0.f;
#pragma unroll
      for (int i = 0; i < 8; ++i) { p += a2[i]; p2 += a2[i] * a2[i]; }
      int c = nt * 16 + (lane & 15);
      atomicAdd(&sCS[c], p);
      atomicAdd(&sCS2[c], p2);
    }
  }
  __syncthreads();
  if (tid < 64) { atomicAdd(&colsum[tid], sCS[tid]); atomicAdd(&colsumsq[tid], sCS2[tid]); }
}

// ---------------- BN finalize ----------------
__global__ void k_bn(const float* colsum, const float* colsumsq, const float* gamma,
                     const float* beta, float* sc, float* sh, int n) {
  int i = threadIdx.x;
  if (i < 64) {
    float inv = 1.f / (float)n;
    float mu = colsum[i] * inv;
    float var = colsumsq[i] * inv - mu * mu;
    float s = rsqrtf(var + 1e-5f) * gamma[i];
    sc[i] = s; sh[i] = beta[i] - mu * s;
  }
}

// ---------------- BN apply + relu + global_add_pool ----------------
__global__ __launch_bounds__(256) void k_pool(const float* __restrict__ out2,
                                              const float* __restrict__ sc,
                                              const float* __restrict__ sh,
                                              const int* __restrict__ batch,
                                              float* __restrict__ pooled, int n) {
  int idx = blockIdx.x * blockDim.x + threadIdx.x;
  if (idx >= n * 64) return;
  int node = idx >> 6, c = idx & 63;
  float v = out2[idx] * sc[c] + sh[c];
  v = fmaxf(v, 0.f);
  atomicAdd(&pooled[(size_t)batch[node] * 64 + c], v);
}

// ---------------- head MLP ----------------
__global__ __launch_bounds__(128) void k_head(const float* __restrict__ pooled,
                                              const float* __restrict__ Wm1,
                                              const float* __restrict__ bm1,
                                              const float* __restrict__ Wm2,
                                              const float* __restrict__ bm2,
                                              float* __restrict__ out) {
  __shared__ float H[64 * 100];
  int tid = threadIdx.x;
  for (int idx = tid; idx < 6400; idx += 128) {
    int b = idx / 100, m = idx - b * 100;
    float h = bm1[m];
    for (int f = 0; f < 64; ++f) h += pooled[b * 64 + f] * Wm1[f * 100 + m];
    H[idx] = fmaxf(h, 0.f);
  }
  __syncthreads();
  if (tid < 64) {
    float o = bm2[0];
    for (int m = 0; m < 100; ++m) o += H[tid * 100 + m] * Wm2[m];
    out[tid] = o;
  }
}

extern "C" void kernel_launch(void* const* d_in, const int* in_sizes, int n_in,
                              void* d_out, int out_size, void* d_ws, size_t ws_size,
                              hipStream_t stream) {
  const float* x         = (const float*)d_in[0];
  const float* edge_attr = (const float*)d_in[1];
  const int*   eidx      = (const int*)d_in[2];
  const int*   batch     = (const int*)d_in[3];
  const float* W1 = (const float*)d_in[4];  const float* b1 = (const float*)d_in[5];
  const float* W2 = (const float*)d_in[6];  const float* b2 = (const float*)d_in[7];
  const float* We = (const float*)d_in[8];  const float* be = (const float*)d_in[9];
  const float* Wp = (const float*)d_in[10]; const float* bp = (const float*)d_in[11];
  const float* Wo = (const float*)d_in[12]; const float* bo = (const float*)d_in[13];
  const float* Wl = (const float*)d_in[14]; const float* bl = (const float*)d_in[15];
  const float* gamma = (const float*)d_in[16]; const float* beta = (const float*)d_in[17];
  const float* Wm1 = (const float*)d_in[18]; const float* bm1 = (const float*)d_in[19];
  const float* Wm2 = (const float*)d_in[20]; const float* bm2 = (const float*)d_in[21];

  int N = in_sizes[0] / 64;
  int E = in_sizes[2] / 2;
  int ntiles = N / 16;  // N=100000 -> 6250

  float* ws = (float*)d_ws;
  size_t N64 = (size_t)N * 64;
  float*    xnew  = ws;
  float*    xd    = ws + N64;
  float*    xs    = ws + 2 * N64;   // reused as out2 after scatter
  float*    sum   = ws + 3 * N64;
  float*    sumsq = ws + 4 * N64;
  unsigned* mne   = (unsigned*)(ws + 5 * N64);
  unsigned* mxe   = (unsigned*)(ws + 6 * N64);
  float*    cnt   = ws + 7 * N64;
  float*    Wec   = cnt + N;
  float*    cvec  = Wec + 1024;
  float*    colsum   = cvec + 64;
  float*    colsumsq = colsum + 64;
  float*    bnsc  = colsumsq + 64;
  float*    bnsh  = bnsc + 64;
  float*    pooled = bnsh + 64;  // 64*64

  int tb = (ntiles + 3) / 4;
  k_prep<<<1, 64, 0, stream>>>(We, be, Wp, Wec, cvec);
  k_init<<<(int)((N64 + 255) / 256), 256, 0, stream>>>(sum, sumsq, mne, mxe, cnt,
                                                       colsum, colsumsq, pooled,
                                                       (int)N64, N);
  k_premlp<<<tb, 128, 0, stream>>>(x, W1, b1, W2, b2, xnew, ntiles);
  k_proj<<<tb, 128, 0, stream>>>(xnew, Wp, xd, xs, ntiles);
  k_scatter<<<(E + 7) / 8, 256, 0, stream>>>(edge_attr, eidx, xs, Wec, cvec,
                                             sum, sumsq, mne, mxe, cnt, E);
  k_node<<<tb, 128, 0, stream>>>(xnew, xd, sum, sumsq, mne, mxe, cnt, bp,
                                 Wo, bo, Wl, bl, xs /*out2*/, colsum, colsumsq, ntiles);
  k_bn<<<1, 64, 0, stream>>>(colsum, colsumsq, gamma, beta, bnsc, bnsh, N);
  k_pool<<<(int)((N64 + 255) / 256), 256, 0, stream>>>(xs, bnsc, bnsh, batch, pooled, N);
  k_head<<<1, 128, 0, stream>>>(pooled, Wm1, bm1, Wm2, bm2, (float*)d_out);
}